// HungarianMatcher_41944650612880
// MI455X (gfx1250) — compile-verified
//
#include <hip/hip_runtime.h>
#include <hip/hip_bf16.h>
#include <math.h>

#define BS  16
#define Q   900
#define NC  91
#define PNC 92     // padded class stride (multiple of 4 for the K-loop)
#define T   100

typedef float v2f __attribute__((ext_vector_type(2)));
typedef float v8f __attribute__((ext_vector_type(8)));

// ---------------------------------------------------------------------------
// Kernel A: row softmax over NC=91 classes, written with padded stride PNC=92
// (pad column zeroed). One wave (32 lanes) per row.
// ---------------------------------------------------------------------------
__global__ __launch_bounds__(32) void softmax_rows(const float* __restrict__ logits,
                                                   float* __restrict__ probs) {
    const int row  = blockIdx.x;            // 0 .. BS*Q-1
    const int lane = threadIdx.x;           // 0 .. 31 (wave32)
    const float* x = logits + (size_t)row * NC;

    float m = -INFINITY;
    for (int c = lane; c < NC; c += 32) m = fmaxf(m, x[c]);
    for (int off = 16; off > 0; off >>= 1) m = fmaxf(m, __shfl_xor(m, off));

    float s = 0.f;
    for (int c = lane; c < NC; c += 32) s += expf(x[c] - m);
    for (int off = 16; off > 0; off >>= 1) s += __shfl_xor(s, off);

    for (int c = lane; c < PNC; c += 32) {
        float pv = 0.f;
        if (c < NC) pv = expf(x[c] - m) / s;
        probs[(size_t)row * PNC + c] = pv;
    }
}

// ---------------------------------------------------------------------------
// Kernel B: per-batch cost matrix, stored transposed Ct[b][t][q].
// cost_class via one-hot GEMM on V_WMMA_F32_16X16X4_F32 (exact in f32):
// 23 unconditional K-steps, each A-fragment a single 8B-aligned b64 load.
// L1(cxcywh) + GIoU fused in the tile epilogue.
// One wave computes one 16x16 (q,t) tile; EXEC stays all-ones through the
// WMMA loop (row index clamped, never masked).
// ---------------------------------------------------------------------------
__global__ __launch_bounds__(32) void cost_tiles(const float* __restrict__ probs,
                                                 const float4* __restrict__ pred_boxes,
                                                 const float4* __restrict__ tgt_boxes,
                                                 const int* __restrict__ tgt_labels,
                                                 float* __restrict__ Ct) {
    const int b     = blockIdx.z;
    const int qtile = blockIdx.x * 16;
    const int ttile = blockIdx.y * 16;
    const int lane  = threadIdx.x;
    const int half  = lane >> 4;            // 0: K pair {0,1}, 1: K pair {2,3}
    const int l15   = lane & 15;

    // A-fragment source row (M = l15); clamp instead of mask (stores are guarded)
    int qa = qtile + l15;
    if (qa >= Q) qa = Q - 1;
    const float2* __restrict__ prow2 =
        (const float2*)(probs + ((size_t)b * Q + qa) * PNC);

    // B-fragment one-hot column (N = l15)
    const int tb  = ttile + l15;
    const int lbl = (tb < T) ? tgt_labels[b * T + tb] : -1;

    v8f acc = {};   // zero accumulator
    #pragma unroll
    for (int kk = 0; kk < PNC / 4; ++kk) {          // 23 WMMA steps
        const int    k0 = 4 * kk + 2 * half;
        const float2 av = prow2[2 * kk + half];     // elements k0, k0+1
        v2f a, bb;
        a.x  = av.x;
        a.y  = av.y;
        bb.x = (lbl == k0)     ? 1.f : 0.f;
        bb.y = (lbl == k0 + 1) ? 1.f : 0.f;
        acc = __builtin_amdgcn_wmma_f32_16x16x4_f32(false, a, false, bb,
                                                    (short)0, acc, false, false);
    }

    // Epilogue: D layout -> lanes 0-15: N=lane, VGPR r: M=r; lanes 16-31: M=r+8
    const int n = l15;
    const int t = ttile + n;
    const int mbase = half * 8;
    float4 ts = (t < T) ? tgt_boxes[b * T + t] : make_float4(0.f, 0.f, 0.f, 0.f);

    const float tcx = (ts.x + ts.z) * 0.5f, tcy = (ts.y + ts.w) * 0.5f;
    const float tw  = ts.z - ts.x,          th  = ts.w - ts.y;
    const float a2  = tw * th;

    #pragma unroll
    for (int r = 0; r < 8; ++r) {
        const int q = qtile + mbase + r;
        if (q < Q && t < T) {
            const float4 ps = pred_boxes[b * Q + q];
            const float pcls = acc[r];
            // L1 on cxcywh
            const float ocx = (ps.x + ps.z) * 0.5f, ocy = (ps.y + ps.w) * 0.5f;
            const float ow  = ps.z - ps.x,          oh  = ps.w - ps.y;
            const float l1  = fabsf(ocx - tcx) + fabsf(ocy - tcy)
                            + fabsf(ow - tw)   + fabsf(oh - th);
            // GIoU on xyxy
            const float a1  = ow * oh;
            const float iw  = fmaxf(fminf(ps.z, ts.z) - fmaxf(ps.x, ts.x), 0.f);
            const float ih  = fmaxf(fminf(ps.w, ts.w) - fmaxf(ps.y, ts.y), 0.f);
            const float inter = iw * ih;
            const float uni   = a1 + a2 - inter;
            const float iou   = inter / (uni + 1e-8f);
            const float Ww = fmaxf(fmaxf(ps.z, ts.z) - fminf(ps.x, ts.x), 0.f);
            const float Wh = fmaxf(fmaxf(ps.w, ts.w) - fminf(ps.y, ts.y), 0.f);
            const float area = Ww * Wh + 1e-8f;
            const float giou = iou - (area - uni) / area;

            Ct[((size_t)b * T + t) * Q + q] = 5.0f * l1 - pcls - 2.0f * giou;
        }
    }
}

// ---------------------------------------------------------------------------
// Kernel C: Jonker-Volgenant LSA per batch (f64 duals, matching reference),
// one 256-thread workgroup per batch; column scans + argmin reduced in LDS.
// Then rank-sort pairs by query index and emit src/tgt/keep as floats.
// ---------------------------------------------------------------------------
#define NT 256
__global__ __launch_bounds__(NT) void hungarian_assign(const float* __restrict__ Ct,
                                                       const float4* __restrict__ pred_boxes,
                                                       const float4* __restrict__ tgt_boxes,
                                                       float* __restrict__ out) {
    const int b   = blockIdx.x;
    const int tid = threadIdx.x;

    __shared__ double sv[Q + 1];
    __shared__ double sminv[Q + 1];
    __shared__ double su[T + 1];
    __shared__ int    sp[Q + 1];
    __shared__ int    sway[Q + 1];
    __shared__ unsigned char sused[Q + 1];
    __shared__ double rval[NT];
    __shared__ int    ridx[NT];
    __shared__ int    s_j0, s_j1;
    __shared__ double s_delta;
    __shared__ int    q_of_t[T];

    const double DINF = 1e300;
    const float* Cb = Ct + (size_t)b * T * Q;

    for (int j = tid; j <= Q; j += NT) { sv[j] = 0.0; sp[j] = 0; sway[j] = 0; }
    for (int i = tid; i <= T; i += NT) su[i] = 0.0;
    __syncthreads();

    for (int i = 1; i <= T; ++i) {
        if (tid == 0) { sp[0] = i; s_j0 = 0; }
        for (int j = tid; j <= Q; j += NT) { sminv[j] = DINF; sused[j] = 0; }
        __syncthreads();

        for (;;) {
            if (tid == 0) sused[s_j0] = 1;
            __syncthreads();
            const int    j0s = s_j0;
            const int    i0  = sp[j0s];
            const double ui0 = su[i0];
            const float* crow = Cb + (size_t)(i0 - 1) * Q;

            double lbest = DINF;
            int    lidx  = Q + 1;
            for (int j = 1 + tid; j <= Q; j += NT) {
                if (!sused[j]) {
                    const double cur = (double)crow[j - 1] - ui0 - sv[j];
                    if (cur < sminv[j]) { sminv[j] = cur; sway[j] = j0s; }
                    const double mv = sminv[j];
                    if (mv < lbest) { lbest = mv; lidx = j; }
                }
            }
            rval[tid] = lbest; ridx[tid] = lidx;
            __syncthreads();
            for (int off = NT / 2; off > 0; off >>= 1) {
                if (tid < off) {
                    const double ov = rval[tid + off];
                    const int    oi = ridx[tid + off];
                    if (ov < rval[tid] || (ov == rval[tid] && oi < ridx[tid])) {
                        rval[tid] = ov; ridx[tid] = oi;
                    }
                }
                __syncthreads();
            }
            if (tid == 0) { s_delta = rval[0]; s_j1 = ridx[0]; }
            __syncthreads();

            const double delta = s_delta;
            for (int j = tid; j <= Q; j += NT) {
                if (sused[j]) { su[sp[j]] += delta; sv[j] -= delta; }
                else          sminv[j] -= delta;
            }
            __syncthreads();
            if (tid == 0) s_j0 = s_j1;
            __syncthreads();
            if (sp[s_j0] == 0) break;
        }
        // augment (serial, <=100 hops)
        if (tid == 0) {
            int j0 = s_j0;
            while (j0 != 0) { const int j1 = sway[j0]; sp[j0] = sp[j1]; j0 = j1; }
        }
        __syncthreads();
    }

    // col->row  =>  per-target assigned query
    for (int j = 1 + tid; j <= Q; j += NT) {
        const int r = sp[j];
        if (r > 0) q_of_t[r - 1] = j - 1;
    }
    __syncthreads();

    // rank-sort by query index (distinct values), emit outputs as floats
    for (int t = tid; t < T; t += NT) {
        const int q = q_of_t[t];
        int rank = 0;
        for (int t2 = 0; t2 < T; ++t2) rank += (q_of_t[t2] < q);

        out[(size_t)b * T + rank]              = (float)q;   // src
        out[(size_t)BS * T + b * T + rank]     = (float)t;   // tgt

        const float4 ps = pred_boxes[b * Q + q];
        const float4 ts = tgt_boxes[b * T + t];
        const float iw = fmaxf(fminf(ps.z, ts.z) - fmaxf(ps.x, ts.x), 0.f);
        const float ih = fmaxf(fminf(ps.w, ts.w) - fmaxf(ps.y, ts.y), 0.f);
        const float inter = iw * ih;
        const float a1 = (ps.z - ps.x) * (ps.w - ps.y);
        const float a2 = (ts.z - ts.x) * (ts.w - ts.y);
        const float iou = inter / (a1 + a2 - inter + 1e-8f);
        out[(size_t)2 * BS * T + b * T + rank] = (iou > 0.5f) ? 1.f : 0.f; // keep
    }
}

// ---------------------------------------------------------------------------
extern "C" void kernel_launch(void* const* d_in, const int* in_sizes, int n_in,
                              void* d_out, int out_size, void* d_ws, size_t ws_size,
                              hipStream_t stream) {
    const float* pred_logits = (const float*)d_in[0];   // (BS,Q,NC) f32
    const float* pred_boxes  = (const float*)d_in[1];   // (BS,Q,4)  f32
    const float* tgt_boxes   = (const float*)d_in[2];   // (BS,T,4)  f32
    const int*   tgt_labels  = (const int*)d_in[3];     // (BS,T)    i32
    float* out = (float*)d_out;                         // [src|tgt|keep] 3*BS*T

    float* probs = (float*)d_ws;                        // BS*Q*PNC floats
    float* Ct    = probs + (size_t)BS * Q * PNC;        // BS*T*Q floats (t-major)

    softmax_rows<<<BS * Q, 32, 0, stream>>>(pred_logits, probs);

    dim3 gridB((Q + 15) / 16, (T + 15) / 16, BS);
    cost_tiles<<<gridB, 32, 0, stream>>>(probs,
                                         (const float4*)pred_boxes,
                                         (const float4*)tgt_boxes,
                                         tgt_labels, Ct);

    hungarian_assign<<<BS, NT, 0, stream>>>(Ct,
                                            (const float4*)pred_boxes,
                                            (const float4*)tgt_boxes,
                                            out);
}